// RelationalKenn_23287312679568
// MI455X (gfx1250) — compile-verified
//
#include <hip/hip_runtime.h>

typedef __attribute__((ext_vector_type(2))) float v2f;
typedef __attribute__((ext_vector_type(8))) float v8f;

#define N_NODES 100000
#define N_U 16
#define N_EDGES 3200000

// ---------------------------------------------------------------------------
// Kernel A: unary knowledge enhancer.
//   sel = Z @ S      (S[p][j] = sign[j] * (u_idx[j]==p), one-hot gather)
//   m   = pairwise softmax over literal slots (j=2c, 2c+1)
//   D   = m @ T      (T[j][p] = w[c]*sign[j]*(u_idx[j]==p), weighted scatter)
//   u   = Z + D
// Each wave32 owns a 16-row tile; both GEMMs are chains of 4x
// V_WMMA_F32_16X16X4_F32. 100000 % 16 == 0, so tiles are exact.
// Table values are pre-loaded unconditionally so fragment build is v_cndmask,
// not predicated loads.
// ---------------------------------------------------------------------------
__global__ __launch_bounds__(256) void kenn_unary_wmma(
    const float* __restrict__ unary,
    const int*   __restrict__ u_idx,    // [8,2] flat (16)
    const float* __restrict__ u_signs,  // [8,2] flat (16)
    const float* __restrict__ w_u,      // [8]
    float*       __restrict__ u_out)    // [N_NODES,16] -> d_out region 0
{
  __shared__ float lds[8 * 16 * 18];    // 16x16 softmax tile per wave, stride-18 pad
  const int wave = threadIdx.x >> 5;
  const int lane = threadIdx.x & 31;
  const int h    = lane >> 4;           // lane half (0/1)
  const int n    = lane & 15;           // row (A) / column (B,C,D) coordinate
  const int tile = blockIdx.x * 8 + wave;
  if (tile * 16 >= N_NODES) return;
  const int row0 = tile * 16;

  // Pre-load tiny clause tables into registers (unconditional -> cndmask).
  const int   my_idx  = u_idx[n];       // literal slot j = n for S fragments
  const float my_sign = u_signs[n];
  int   tj [8];                         // T-fragment rows j = 4s + 2h + {0,1}
  float tv [8];
#pragma unroll
  for (int s = 0; s < 4; ++s) {
    const int j0 = 4 * s + 2 * h;
    const int j1 = j0 + 1;
    tj[2 * s]     = u_idx[j0];
    tj[2 * s + 1] = u_idx[j1];
    tv[2 * s]     = w_u[j0 >> 1] * u_signs[j0];
    tv[2 * s + 1] = w_u[j1 >> 1] * u_signs[j1];
  }

  // ---- sel = Z @ S -------------------------------------------------------
  v8f acc = {};
#pragma unroll
  for (int s = 0; s < 4; ++s) {
    const int k0 = 4 * s + 2 * h;       // K-slice columns for this lane half
    v2f a;                              // A: Z[row = n][k0], Z[row = n][k0+1]
    const float2 za = *(const float2*)(unary + (size_t)(row0 + n) * N_U + k0);
    a.x = za.x; a.y = za.y;
    v2f b;                              // B: S[p = k0 / k0+1][j = n]
    b.x = (my_idx == k0)     ? my_sign : 0.0f;
    b.y = (my_idx == k0 + 1) ? my_sign : 0.0f;
    acc = __builtin_amdgcn_wmma_f32_16x16x4_f32(false, a, false, b,
                                                (short)0, acc, false, false);
  }

  // ---- pairwise softmax (column j = n; partner column n^1 is lane^1) -----
  float* mt = &lds[wave * 16 * 18];
#pragma unroll
  for (int r = 0; r < 8; ++r) {
    const float self = acc[r];
    const float part = __shfl_xor(self, 1);
    const float m    = 1.0f / (1.0f + __expf(part - self)); // e^s/(e^s+e^p)
    mt[(r + 8 * h) * 18 + n] = m;       // row-major, padded stride
  }
  asm volatile("s_wait_dscnt 0" ::: "memory");  // intra-wave LDS RAW order

  // ---- D = m @ T ---------------------------------------------------------
  v8f d = {};
#pragma unroll
  for (int s = 0; s < 4; ++s) {
    const int j0 = 4 * s + 2 * h;       // literal-slot rows of T
    v2f a;                              // A: m[row = n][j0], m[row = n][j0+1]
    const float2 ma = *(const float2*)(mt + n * 18 + j0);
    a.x = ma.x; a.y = ma.y;
    v2f b;                              // B: T[j0 / j0+1][p = n]
    b.x = (tj[2 * s]     == n) ? tv[2 * s]     : 0.0f;
    b.y = (tj[2 * s + 1] == n) ? tv[2 * s + 1] : 0.0f;
    d = __builtin_amdgcn_wmma_f32_16x16x4_f32(false, a, false, b,
                                              (short)0, d, false, false);
  }

  // ---- u = Z + D (C layout: lane(h,n) holds rows r+8h, column n) ---------
#pragma unroll
  for (int r = 0; r < 8; ++r) {
    const int row = row0 + r + 8 * h;
    u_out[(size_t)row * N_U + n] = unary[(size_t)row * N_U + n] + d[r];
  }
}

// ---------------------------------------------------------------------------
// Kernel B1: deterministic ".set" winners — highest edge id wins per node.
// ---------------------------------------------------------------------------
__global__ __launch_bounds__(256) void kenn_winners(
    const int* __restrict__ index1, const int* __restrict__ index2,
    int* __restrict__ win1, int* __restrict__ win2)
{
  const int e = blockIdx.x * 256 + threadIdx.x;   // grid sized exactly
  atomicMax(&win1[index1[e]], e);
  atomicMax(&win2[index2[e]], e);
}

// ---------------------------------------------------------------------------
// Kernel B2: binary clause enhancer. 16 lanes per edge -> every node-row
// gather/scatter is one contiguous 64B transaction. Clause softmax deltas
// accumulate in LDS (ds_add_f32), then the winning edge scatters its delta.
// 3.2M % 16 == 0, so no tail guard (block-uniform barriers).
// ---------------------------------------------------------------------------
__global__ __launch_bounds__(256) void kenn_edges(
    const float* __restrict__ u,        // enhanced unary (d_out region 0)
    const float* __restrict__ binary,
    const int*   __restrict__ index1,
    const int*   __restrict__ index2,
    const int*   __restrict__ b_idx,    // [8,3] flat (24)
    const float* __restrict__ b_signs,  // [8,3] flat (24)
    const float* __restrict__ w_b,      // [8]
    const int*   __restrict__ win1,
    const int*   __restrict__ win2,
    float*       __restrict__ bufA,     // [N_NODES,16] zero-initialized
    float*       __restrict__ bufB,     // [N_NODES,16] zero-initialized
    float*       __restrict__ bin_out)  // [N_EDGES]
{
  __shared__ float zb[16][36];          // joined vector z[33] per edge slot
  __shared__ float db[16][36];          // delta accumulator per edge slot
  const int el = threadIdx.x >> 4;      // edge slot within block
  const int t  = threadIdx.x & 15;      // cooperating lane within edge
  const int e  = blockIdx.x * 16 + el;
  const int n1 = index1[e];
  const int n2 = index2[e];

  zb[el][t]      = u[(size_t)n1 * N_U + t];       // 64B coalesced gather
  zb[el][16 + t] = u[(size_t)n2 * N_U + t];
  db[el][t]      = 0.0f;
  db[el][16 + t] = 0.0f;
  if (t == 0) { zb[el][32] = binary[e]; db[el][32] = 0.0f; }
  __syncthreads();

  if (t < 8) {                          // one clause per lane
    const int   c  = t;
    const int   j0 = b_idx[c * 3 + 0], j1 = b_idx[c * 3 + 1], j2 = b_idx[c * 3 + 2];
    const float s0 = b_signs[c * 3 + 0], s1 = b_signs[c * 3 + 1], s2 = b_signs[c * 3 + 2];
    const float x0 = s0 * zb[el][j0];
    const float x1 = s1 * zb[el][j1];
    const float x2 = s2 * zb[el][j2];
    const float mx = fmaxf(x0, fmaxf(x1, x2));
    const float e0 = __expf(x0 - mx), e1 = __expf(x1 - mx), e2 = __expf(x2 - mx);
    const float wi = w_b[c] / (e0 + e1 + e2);
    atomicAdd(&db[el][j0], wi * e0 * s0);
    atomicAdd(&db[el][j1], wi * e1 * s1);
    atomicAdd(&db[el][j2], wi * e2 * s2);
  }
  __syncthreads();

  if (win1[n1] == e) bufA[(size_t)n1 * N_U + t] = db[el][t];        // .set
  if (win2[n2] == e) bufB[(size_t)n2 * N_U + t] = db[el][16 + t];   // .set
  if (t == 0) bin_out[e] = zb[el][32] + db[el][32];
}

// ---------------------------------------------------------------------------
// Kernel C: out_u = u + delta_up (bufA/bufB default 0 for untouched nodes).
// ---------------------------------------------------------------------------
__global__ __launch_bounds__(256) void kenn_finalize(
    float* __restrict__ out_u,
    const float* __restrict__ bufA,
    const float* __restrict__ bufB)
{
  const int i = blockIdx.x * 256 + threadIdx.x;   // grid sized exactly
  out_u[i] = out_u[i] + bufA[i] + bufB[i];
}

// ---------------------------------------------------------------------------
extern "C" void kernel_launch(void* const* d_in, const int* in_sizes, int n_in,
                              void* d_out, int out_size, void* d_ws, size_t ws_size,
                              hipStream_t stream) {
  const float* unary   = (const float*)d_in[0];
  const float* binary  = (const float*)d_in[1];
  const int*   index1  = (const int*)  d_in[2];
  const int*   index2  = (const int*)  d_in[3];
  const int*   u_idx   = (const int*)  d_in[4];
  const float* u_signs = (const float*)d_in[5];
  const int*   b_idx   = (const int*)  d_in[6];
  const float* b_signs = (const float*)d_in[7];
  const float* w_u     = (const float*)d_in[8];
  const float* w_b     = (const float*)d_in[9];

  float* out_u = (float*)d_out;                       // [N_NODES*16]
  float* out_b = out_u + (size_t)N_NODES * N_U;       // [N_EDGES]

  // Workspace layout: bufA | bufB | win1 | win2  (~13.6 MB total)
  float* bufA = (float*)d_ws;
  float* bufB = bufA + (size_t)N_NODES * N_U;
  int*   win1 = (int*)(bufB + (size_t)N_NODES * N_U);
  int*   win2 = win1 + N_NODES;

  hipMemsetAsync(bufA, 0,    (size_t)N_NODES * N_U * 2 * sizeof(float), stream);
  hipMemsetAsync(win1, 0xFF, (size_t)N_NODES * 2 * sizeof(int), stream);

  kenn_unary_wmma<<<(N_NODES / 16 + 7) / 8, 256, 0, stream>>>(
      unary, u_idx, u_signs, w_u, out_u);
  kenn_winners<<<N_EDGES / 256, 256, 0, stream>>>(index1, index2, win1, win2);
  kenn_edges<<<N_EDGES / 16, 256, 0, stream>>>(
      out_u, binary, index1, index2, b_idx, b_signs, w_b, win1, win2,
      bufA, bufB, out_b);
  kenn_finalize<<<(N_NODES * N_U) / 256, 256, 0, stream>>>(out_u, bufA, bufB);
}